// BinnedLoss_74491912782169
// MI455X (gfx1250) — compile-verified
//
#include <hip/hip_runtime.h>
#include <hip/hip_bf16.h>
#include <math.h>

#define BINS      128
#define TPB       256
#define NWAVES    (TPB / 32)      // 8 waves per block (wave32)
#define SUBSTRIDE 132             // padded per-wave histogram stride
#define NBLOCKS   2048

typedef __attribute__((ext_vector_type(2))) float v2f;
typedef __attribute__((ext_vector_type(8))) float v8f;

// ---- order-preserving float <-> unsigned key (for integer atomic min/max) ----
__device__ __forceinline__ unsigned orderedKey(float f) {
    unsigned u = __float_as_uint(f);
    return (u & 0x80000000u) ? ~u : (u | 0x80000000u);
}
__device__ __forceinline__ float keyToFloat(unsigned k) {
    unsigned u = (k & 0x80000000u) ? (k & 0x7FFFFFFFu) : ~k;
    return __uint_as_float(u);
}

// ---------------- workspace layout (as unsigned/float words) ----------------
// [0] min key   [1] max key   [2 .. 129] Hs accum   [130 .. 257] He accum
__global__ void init_ws_kernel(unsigned* wsu) {
    int t = threadIdx.x;
    if (t == 0) wsu[0] = 0xFFFFFFFFu;           // min identity
    if (t == 1) wsu[1] = 0x00000000u;           // max identity
    if (t >= 2 && t < 2 + 2 * BINS) ((float*)wsu)[t] = 0.0f;
}

// ---------------------------- pass 1: min / max -----------------------------
__global__ void minmax_kernel(const float* __restrict__ sim,
                              const float* __restrict__ gexp,
                              long long n, unsigned* wsu) {
    long long tid    = (long long)blockIdx.x * blockDim.x + threadIdx.x;
    long long stride = (long long)gridDim.x * blockDim.x;
    const float4* s4 = (const float4*)sim;
    const float4* e4 = (const float4*)gexp;
    long long n4 = n >> 2;

    float lmin = INFINITY, lmax = -INFINITY;
    for (long long i = tid; i < n4; i += stride) {
        __builtin_prefetch((const void*)(s4 + i + stride), 0, 0);
        __builtin_prefetch((const void*)(e4 + i + stride), 0, 0);
        float4 a = s4[i], b = e4[i];
        lmin = fminf(lmin, fminf(fminf(a.x, a.y), fminf(a.z, a.w)));
        lmin = fminf(lmin, fminf(fminf(b.x, b.y), fminf(b.z, b.w)));
        lmax = fmaxf(lmax, fmaxf(fmaxf(a.x, a.y), fmaxf(a.z, a.w)));
        lmax = fmaxf(lmax, fmaxf(fmaxf(b.x, b.y), fmaxf(b.z, b.w)));
    }
    for (long long i = (n4 << 2) + tid; i < n; i += stride) {   // tail
        float a = sim[i], b = gexp[i];
        lmin = fminf(lmin, fminf(a, b));
        lmax = fmaxf(lmax, fmaxf(a, b));
    }
    // wave32 reduction
    for (int off = 16; off > 0; off >>= 1) {
        lmin = fminf(lmin, __shfl_xor(lmin, off, 32));
        lmax = fmaxf(lmax, __shfl_xor(lmax, off, 32));
    }
    __shared__ float smin[NWAVES], smax[NWAVES];
    int lane = threadIdx.x & 31, wid = threadIdx.x >> 5;
    if (lane == 0) { smin[wid] = lmin; smax[wid] = lmax; }
    __syncthreads();
    if (threadIdx.x == 0) {
        float bmin = smin[0], bmax = smax[0];
        for (int i = 1; i < NWAVES; ++i) {
            bmin = fminf(bmin, smin[i]);
            bmax = fmaxf(bmax, smax[i]);
        }
        atomicMin(&wsu[0], orderedKey(bmin));
        atomicMax(&wsu[1], orderedKey(bmax));
    }
}

// ------------------------- pass 2: tent histograms --------------------------
// Bin with a hoisted reciprocal (no per-element v_div chain), then repair any
// off-by-one from reciprocal rounding branchlessly against the exact interval
// test hk <= x < hk1, which is what the reference's in_iv mask enforces.
__device__ __forceinline__ void deposit(float x, float w, float mn, float step,
                                        float inv_step, float* __restrict__ h) {
    int k  = (int)floorf((x - mn) * inv_step);
    int kc = min(max(k, 0), BINS - 2);
    float hk  = mn + (float)kc * step;
    float hk1 = mn + (float)(kc + 1) * step;
    int adj = (x >= hk1) - (x < hk);               // -1 / 0 / +1 correction
    if (adj != 0) {
        kc  = min(max(kc + adj, 0), BINS - 2);
        hk  = mn + (float)kc * step;
        hk1 = mn + (float)(kc + 1) * step;
    }
    if (x >= hk && x < hk1) {                      // strict interval mask
        if (kc <= BINS - 3) atomicAdd(&h[kc + 1], w * (x - hk));   // ds_add_f32
        if (kc >= 1)        atomicAdd(&h[kc],     w * (hk1 - x));  // ds_add_f32
    }
}

__global__ void hist_kernel(const float* __restrict__ sim,
                            const float* __restrict__ gexp,
                            const float* __restrict__ wgt,
                            long long n, const unsigned* __restrict__ wsu,
                            float* __restrict__ gHs, float* __restrict__ gHe) {
    __shared__ float lh[2 * NWAVES * SUBSTRIDE];     // 8 sub-hists x 2 = 8.25 KB
    float* hs = lh;
    float* he = lh + NWAVES * SUBSTRIDE;
    for (int i = threadIdx.x; i < 2 * NWAVES * SUBSTRIDE; i += blockDim.x) lh[i] = 0.0f;
    __syncthreads();

    float mn = keyToFloat(wsu[0]);
    float mx = keyToFloat(wsu[1]);
    float step = (mx - mn) / (float)(BINS - 1);
    float inv_step = 1.0f / step;                    // one IEEE div per thread

    int sub = threadIdx.x >> 5;                      // one sub-histogram per wave
    float* hsw = hs + sub * SUBSTRIDE;
    float* hew = he + sub * SUBSTRIDE;

    long long tid    = (long long)blockIdx.x * blockDim.x + threadIdx.x;
    long long stride = (long long)gridDim.x * blockDim.x;
    const float4* s4 = (const float4*)sim;
    const float4* e4 = (const float4*)gexp;
    const float4* w4 = (const float4*)wgt;
    long long n4 = n >> 2;

    for (long long i = tid; i < n4; i += stride) {
        __builtin_prefetch((const void*)(s4 + i + stride), 0, 0);  // global_prefetch_b8
        __builtin_prefetch((const void*)(e4 + i + stride), 0, 0);
        __builtin_prefetch((const void*)(w4 + i + stride), 0, 0);
        float4 a = s4[i], b = e4[i], ww = w4[i];
        deposit(a.x, ww.x, mn, step, inv_step, hsw);
        deposit(a.y, ww.y, mn, step, inv_step, hsw);
        deposit(a.z, ww.z, mn, step, inv_step, hsw);
        deposit(a.w, ww.w, mn, step, inv_step, hsw);
        deposit(b.x, 1.0f, mn, step, inv_step, hew);
        deposit(b.y, 1.0f, mn, step, inv_step, hew);
        deposit(b.z, 1.0f, mn, step, inv_step, hew);
        deposit(b.w, 1.0f, mn, step, inv_step, hew);
    }
    for (long long i = (n4 << 2) + tid; i < n; i += stride) {      // tail
        deposit(sim[i],  wgt[i], mn, step, inv_step, hsw);
        deposit(gexp[i], 1.0f,   mn, step, inv_step, hew);
    }
    __syncthreads();

    // fold 8 sub-histograms, one global float atomic per (hist,bin) per block
    for (int b = threadIdx.x; b < 2 * BINS; b += blockDim.x) {
        int bin = b & (BINS - 1);
        const float* src = (b < BINS) ? hs : he;
        float acc = 0.0f;
        for (int c = 0; c < NWAVES; ++c) acc += src[c * SUBSTRIDE + bin];
        atomicAdd(((b < BINS) ? gHs : gHe) + bin, acc);            // global_atomic_add_f32
    }
}

// ---------------- pass 3: normalize + L2, single wave32 ---------------------
// Exact-f32 wave sum via V_WMMA_F32_16X16X4_F32 with B = ones:
// D[m][n] = sum_k A[m][k]; lane-local sum of the 8 D VGPRs covers half the
// rows of one column, shfl_xor(16) adds the other half -> full 32-lane sum.
// Layout-independent since B is all ones.
__device__ __forceinline__ float wave_sum_wmma(float x) {
    v2f a; a.x = x;    a.y = 0.0f;
    v2f b; b.x = 1.0f; b.y = 1.0f;
    v8f c = {};
    c = __builtin_amdgcn_wmma_f32_16x16x4_f32(false, a, false, b,
                                              (short)0, c, false, false);
    float s = c[0] + c[1] + c[2] + c[3] + c[4] + c[5] + c[6] + c[7];
    s += __shfl_xor(s, 16, 32);
    return s;
}

__global__ void final_kernel(const unsigned* __restrict__ wsu,
                             const float* __restrict__ gHs,
                             const float* __restrict__ gHe,
                             float* __restrict__ out) {
    int lane = threadIdx.x;                      // exactly 32 threads, EXEC all-1s
    float mn = keyToFloat(wsu[0]);
    float mx = keyToFloat(wsu[1]);
    float delta = (mx - mn) / (float)BINS;

    float hsv[4], hev[4];
    float ls = 0.0f, le = 0.0f;
    for (int j = 0; j < 4; ++j) {
        hsv[j] = gHs[lane * 4 + j];
        hev[j] = gHe[lane * 4 + j];
        ls += hsv[j];
        le += hev[j];
    }
    float sumS = wave_sum_wmma(ls);
    float sumE = wave_sum_wmma(le);
    float dS = sumS * delta, dE = sumE * delta;

    float acc = 0.0f;
    for (int j = 0; j < 4; ++j) {
        float d = hsv[j] / dS - hev[j] / dE;
        acc += d * d;
    }
    float loss = wave_sum_wmma(acc);
    if (lane == 0) out[0] = loss;
}

// ------------------------------- launcher -----------------------------------
extern "C" void kernel_launch(void* const* d_in, const int* in_sizes, int n_in,
                              void* d_out, int out_size, void* d_ws, size_t ws_size,
                              hipStream_t stream) {
    const float* sim  = (const float*)d_in[0];
    const float* gexp = (const float*)d_in[1];
    const float* wgt  = (const float*)d_in[2];
    long long n = (long long)in_sizes[0];

    unsigned* wsu = (unsigned*)d_ws;
    float* gHs = (float*)d_ws + 2;
    float* gHe = gHs + BINS;
    float* out = (float*)d_out;

    init_ws_kernel<<<1, 512, 0, stream>>>(wsu);
    minmax_kernel<<<NBLOCKS, TPB, 0, stream>>>(sim, gexp, n, wsu);
    hist_kernel<<<NBLOCKS, TPB, 0, stream>>>(sim, gexp, wgt, n, wsu, gHs, gHe);
    final_kernel<<<1, 32, 0, stream>>>(wsu, gHs, gHe, out);
}